// DualMSLoss_FineGrained_56453050138920
// MI455X (gfx1250) — compile-verified
//
#include <hip/hip_runtime.h>
#include <hip/hip_bf16.h>
#include <math.h>
#include <stdint.h>

// ---------------------------------------------------------------------------
// DualMSLoss fused pipeline for MI455X (gfx1250, wave32)
//   A[i, 0:512]    = l2_norm(x_contactless[i])
//   A[i, 512:1024] = (1/36) * sum_t normalize(x_cl_tokens[i,t])
//   B[j, *]        = same for contactbased side
//   w  = A @ B^T       (K = 1024, fp32 WMMA 16x16x4, TDM-staged LDS tiles)
//   loss = ms_sample(w) + ms_sample(w^T)
// ---------------------------------------------------------------------------

typedef __attribute__((ext_vector_type(2))) float v2f;
typedef __attribute__((ext_vector_type(8))) float v8f;
typedef __attribute__((ext_vector_type(4))) unsigned int u32x4;
typedef __attribute__((ext_vector_type(8))) unsigned int u32x8;

#define NB   1024   // batch
#define DD   512    // feature dim
#define TT   36     // tokens
#define KD   1024   // concat K

// -------------------- wave32 / block reductions ----------------------------
__device__ __forceinline__ float waveSum(float v) {
#pragma unroll
  for (int off = 16; off > 0; off >>= 1) v += __shfl_xor(v, off, 32);
  return v;
}
__device__ __forceinline__ float waveMin(float v) {
#pragma unroll
  for (int off = 16; off > 0; off >>= 1) v = fminf(v, __shfl_xor(v, off, 32));
  return v;
}
__device__ __forceinline__ float waveMax(float v) {
#pragma unroll
  for (int off = 16; off > 0; off >>= 1) v = fmaxf(v, __shfl_xor(v, off, 32));
  return v;
}

// 256 threads = 8 wave32s; returns the block-wide result to every thread.
__device__ __forceinline__ float blockSum(float v, float* lds) {
  v = waveSum(v);
  if ((threadIdx.x & 31) == 0) lds[threadIdx.x >> 5] = v;
  __syncthreads();
  float s = 0.f;
#pragma unroll
  for (int i = 0; i < 8; ++i) s += lds[i];
  __syncthreads();
  return s;
}
__device__ __forceinline__ float blockMin(float v, float* lds) {
  v = waveMin(v);
  if ((threadIdx.x & 31) == 0) lds[threadIdx.x >> 5] = v;
  __syncthreads();
  float s = 1e30f;
#pragma unroll
  for (int i = 0; i < 8; ++i) s = fminf(s, lds[i]);
  __syncthreads();
  return s;
}
__device__ __forceinline__ float blockMax(float v, float* lds) {
  v = waveMax(v);
  if ((threadIdx.x & 31) == 0) lds[threadIdx.x >> 5] = v;
  __syncthreads();
  float s = -1e30f;
#pragma unroll
  for (int i = 0; i < 8; ++i) s = fmaxf(s, lds[i]);
  __syncthreads();
  return s;
}

// -------------------- Tensor Data Mover: 2D fp32 tile -> LDS ---------------
// Builds the D# per CDNA5 ISA 8.3/8.4 and issues tensor_load_to_lds with the
// 2-SGPR-group form (VADDR2/3 NULL => tensors up to 2D).
//   tile: tile_rows x 32 fp32, row stride KD elements in memory.
//   LDS : rows padded 32->36 DWORDs via pad_interval=32dw / pad_amount=4dw.
__device__ __forceinline__ void tdm_load_tile_f32(uint32_t lds_byte_off,
                                                  const float* gptr,
                                                  uint32_t tile_rows) {
  const uint64_t ga = (uint64_t)(uintptr_t)gptr;
  u32x4 g0;
  g0.x = 0x1u;                                   // count=1, user descriptor
  g0.y = lds_byte_off;                           // lds_addr[63:32]
  g0.z = (uint32_t)ga;                           // global_addr[95:64]
  g0.w = (uint32_t)(ga >> 32) | (2u << 30);      // global_addr[120:96] | type=2
  u32x8 g1;
  g1.s0 = (2u << 16)      // data_size = 4B
        | (1u << 20)      // pad_enable
        | (4u << 22)      // pad_interval: 32 DWORDs
        | (3u << 25);     // pad_amount : 4 DWORDs  (LDS row stride 36 dw)
  g1.s1 = (KD & 0xFFFFu) << 16;                  // tensor_dim0[15:0]
  g1.s2 = (KD >> 16) | ((NB & 0xFFFFu) << 16);   // tensor_dim0[31:16] | dim1 lo
  g1.s3 = (NB >> 16) | (32u << 16);              // tensor_dim1[31:16] | tile_dim0=32
  g1.s4 = tile_rows & 0xFFFFu;                   // tile_dim1 | tile_dim2=0
  g1.s5 = KD;                                    // tensor_dim0_stride[31:0]
  g1.s6 = 0u;                                    // stride hi | dim1_stride lo
  g1.s7 = 0u;
  asm volatile("tensor_load_to_lds %0, %1" :: "s"(g0), "s"(g1) : "memory");
}

// -------------------- kernel: zero the scalar output -----------------------
__global__ void k_zero(float* out) { out[0] = 0.0f; }

// -------------------- kernel: l2-normalized vector half --------------------
// grid (1024, 2), block 256. Thread t owns dims {t, t+256}.
__global__ void __launch_bounds__(256)
k_prep_vecs(const float* __restrict__ xcl, const float* __restrict__ xcb,
            float* __restrict__ Amat, float* __restrict__ Bmat) {
  __shared__ float lds[8];
  const int b = blockIdx.x;
  const float* src = blockIdx.y ? xcb : xcl;
  float* dst       = blockIdx.y ? Bmat : Amat;
  const int t = threadIdx.x;
  const float v0 = src[(size_t)b * DD + t];
  const float v1 = src[(size_t)b * DD + t + 256];
  const float s  = blockSum(v0 * v0 + v1 * v1, lds);
  const float inv = 1.0f / sqrtf(s + 1e-12f);   // matches l2_norm
  dst[(size_t)b * KD + t]       = v0 * inv;
  dst[(size_t)b * KD + t + 256] = v1 * inv;
}

// -------------------- kernel: token normalize + reduce (HBM-bound) ---------
// grid (1024, 2), block 256 = 8 wave32s. Wave-per-token: each wave owns a
// whole 2KB token (4x b128 per lane), reduces ||t|| with shuffles only (zero
// barriers in the streaming loop), accumulates normalized partial sums, then
// one LDS pass merges the 8 wave partials.
__global__ void __launch_bounds__(256)
k_prep_tokens(const float* __restrict__ tcl, const float* __restrict__ tcb,
              float* __restrict__ Amat, float* __restrict__ Bmat) {
  __shared__ __align__(16) float lbuf[8 * DD];   // 16 KB
  const int b = blockIdx.x;
  const float* src = blockIdx.y ? tcb : tcl;
  float* dst       = blockIdx.y ? Bmat : Amat;
  const int wave = threadIdx.x >> 5;
  const int lane = threadIdx.x & 31;

  float4 acc[4];
#pragma unroll
  for (int q = 0; q < 4; ++q) acc[q] = make_float4(0.f, 0.f, 0.f, 0.f);

  for (int tok = wave; tok < TT; tok += 8) {
    const float* row = src + ((size_t)b * TT + tok) * DD;
    float4 v[4];
#pragma unroll
    for (int q = 0; q < 4; ++q)               // contiguous 512B per b128 wave op
      v[q] = *(const float4*)(row + q * 128 + lane * 4);
    float ss = 0.f;
#pragma unroll
    for (int q = 0; q < 4; ++q)
      ss += v[q].x * v[q].x + v[q].y * v[q].y + v[q].z * v[q].z + v[q].w * v[q].w;
    ss = waveSum(ss);
    const float inv = 1.0f / fmaxf(sqrtf(ss), 1e-12f);  // F.normalize semantics
#pragma unroll
    for (int q = 0; q < 4; ++q) {
      acc[q].x += v[q].x * inv; acc[q].y += v[q].y * inv;
      acc[q].z += v[q].z * inv; acc[q].w += v[q].w * inv;
    }
  }
#pragma unroll
  for (int q = 0; q < 4; ++q)
    *(float4*)(lbuf + wave * DD + q * 128 + lane * 4) = acc[q];
  __syncthreads();

  const int t = threadIdx.x;
  float s0 = 0.f, s1 = 0.f;
#pragma unroll
  for (int w = 0; w < 8; ++w) {
    s0 += lbuf[w * DD + t];
    s1 += lbuf[w * DD + t + 256];
  }
  const float sc = 1.0f / 36.0f;  // split of 1/(T*T) onto each side
  dst[(size_t)b * KD + DD + t]       = s0 * sc;
  dst[(size_t)b * KD + DD + t + 256] = s1 * sc;
}

// -------------------- kernel: fp32 WMMA GEMM  w = A @ B^T ------------------
// Block tile 64(M) x 32(N), K stepped by 32. A/B tiles DMA'd into LDS by the
// Tensor Data Mover (one wave issues, TENSORcnt-synced, barrier-published).
// 8 waves/block, each owns one 16x16 fp32 accumulator (8 VGPRs).
#define LDS_STRIDE 36  // 32 + 4 pad (matches TDM pad_interval/pad_amount)
__global__ void __launch_bounds__(256)
k_gemm_wmma(const float* __restrict__ Amat, const float* __restrict__ Bmat,
            float* __restrict__ W, float* __restrict__ WT) {
  __shared__ __align__(16) float As[64 * LDS_STRIDE];
  __shared__ __align__(16) float Bs[32 * LDS_STRIDE];
  const int tid  = threadIdx.x;
  const int wave = tid >> 5;
  const int lane = tid & 31;
  const int m_blk = blockIdx.x * 64;
  const int n_blk = blockIdx.y * 32;
  const int mw = (wave & 3) * 16;   // wave's M offset within block tile
  const int nw = (wave >> 2) * 16;  // wave's N offset within block tile

  const uint32_t as_off = (uint32_t)(size_t)&As[0];
  const uint32_t bs_off = (uint32_t)(size_t)&Bs[0];

  v8f acc = (v8f)(0.0f);

  // WMMA fp32 16x16x4 fragment coordinates (ISA 7.12.2 layouts):
  //   A 16x4 : lane -> M = lane&15, VGPR{0,1} hold K = 2*(lane>>4) + {0,1}
  //   B 4x16 : lane -> N = lane&15, VGPR{0,1} hold K = 2*(lane>>4) + {0,1}
  const int fm = lane & 15;
  const int fk = (lane >> 4) * 2;

  for (int k0 = 0; k0 < KD; k0 += 32) {
    __syncthreads();  // previous iteration's LDS reads complete
    if (wave == 0) {  // one wave drives the TDM; EXEC is ignored by tensor ops
      tdm_load_tile_f32(as_off, Amat + (size_t)m_blk * KD + k0, 64u);
      tdm_load_tile_f32(bs_off, Bmat + (size_t)n_blk * KD + k0, 32u);
      __builtin_amdgcn_s_wait_tensorcnt(0);
    }
    __syncthreads();  // tiles visible to all 8 waves

#pragma unroll
    for (int kk = 0; kk < 32; kk += 4) {
      const v2f af = *(const v2f*)(As + (mw + fm) * LDS_STRIDE + kk + fk);
      const v2f bf = *(const v2f*)(Bs + (nw + fm) * LDS_STRIDE + kk + fk);
      // D = A(16x4) * B(4x16) + C   -> v_wmma_f32_16x16x4_f32
      acc = __builtin_amdgcn_wmma_f32_16x16x4_f32(
          /*neg_a=*/false, af, /*neg_b=*/false, bf,
          /*c_mod=*/(short)0, acc, /*reuse_a=*/false, /*reuse_b=*/false);
    }
  }

  // C/D 16x16 fp32 layout: VGPR r -> M = r + 8*(lane>>4), N = lane&15
  const int cn = lane & 15;
  const int cm = (lane >> 4) * 8;
#pragma unroll
  for (int r = 0; r < 8; ++r) {
    const int gi = m_blk + mw + cm + r;
    const int gj = n_blk + nw + cn;
    const float v = acc[r];
    W[(size_t)gi * NB + gj]  = v;
    WT[(size_t)gj * NB + gi] = v;
  }
}

// -------------------- kernel: multi-similarity loss over rows --------------
// One block (256 threads) per row; thread t owns columns {t, t+256, t+512, t+768}.
__global__ void __launch_bounds__(256)
k_ms_loss(const float* __restrict__ W, const int* __restrict__ labels,
          float* __restrict__ out) {
  __shared__ float lds[8];
  const int i  = blockIdx.x;
  const int t  = threadIdx.x;
  const int li = labels[i];

  float sim[4];
  bool  pos[4];
  float minP = 1e16f, maxN = -1e16f;
#pragma unroll
  for (int q = 0; q < 4; ++q) {
    const int j   = t + q * 256;
    const float s = W[(size_t)i * NB + j];
    const bool  p = (labels[j] == li);
    sim[q] = s;
    pos[q] = p;
    minP = fminf(minP, p ? s : 1e16f);
    maxN = fmaxf(maxN, p ? -1e16f : s);
  }
  minP = blockMin(minP, lds);
  maxN = blockMax(maxN, lds);

  float hp = 0.f, hn = 0.f;
#pragma unroll
  for (int q = 0; q < 4; ++q) {
    const float s = sim[q];
    if (pos[q]) {
      if (s - 0.1f < maxN) hp += expf(-2.0f * (s - 0.5f));
    } else {
      if (s + 0.1f > minP) hn += expf(40.0f * (s - 0.5f));
    }
  }
  hp = blockSum(hp, lds);
  hn = blockSum(hn, lds);
  if (t == 0)
    atomicAdd(out, log1pf(hp) * (1.0f / 2.0f) + log1pf(hn) * (1.0f / 40.0f));
}

// ---------------------------------------------------------------------------
extern "C" void kernel_launch(void* const* d_in, const int* in_sizes, int n_in,
                              void* d_out, int out_size, void* d_ws, size_t ws_size,
                              hipStream_t stream) {
  const float* xcl    = (const float*)d_in[0];
  const float* xcb    = (const float*)d_in[1];
  const float* tcl    = (const float*)d_in[2];
  const float* tcb    = (const float*)d_in[3];
  const int*   labels = (const int*)d_in[4];
  float* out = (float*)d_out;

  float* Amat = (float*)d_ws;                 // [1024 x 1024]
  float* Bmat = Amat + (size_t)NB * KD;       // [1024 x 1024]
  float* W    = Bmat + (size_t)NB * KD;       // [1024 x 1024]
  float* WT   = W    + (size_t)NB * NB;       // [1024 x 1024]

  k_zero<<<1, 1, 0, stream>>>(out);
  k_prep_vecs<<<dim3(NB, 2), 256, 0, stream>>>(xcl, xcb, Amat, Bmat);
  k_prep_tokens<<<dim3(NB, 2), 256, 0, stream>>>(tcl, tcb, Amat, Bmat);
  k_gemm_wmma<<<dim3(NB / 64, NB / 32), 256, 0, stream>>>(Amat, Bmat, W, WT);
  k_ms_loss<<<NB, 256, 0, stream>>>(W, labels, out);
  k_ms_loss<<<NB, 256, 0, stream>>>(WT, labels, out);
}